// SubQAttention_3719441678494
// MI455X (gfx1250) — compile-verified
//
#include <hip/hip_runtime.h>
#include <hip/hip_bf16.h>

// CDNA5 / gfx1250, wave32. Matrix work on V_WMMA_F32_16X16X4_F32 (fp32-exact,
// matches the fp32 reference; D=64 => 16 chained K=4 WMMAs per 16x16 tile).

typedef __attribute__((ext_vector_type(2))) float v2f;
typedef __attribute__((ext_vector_type(8))) float v8f;

#define BB 4
#define HH 16
#define TT 1024
#define DD 64
#define NBINS 16
#define RH 128
#define THRESH 0.05f

// ---------------------------------------------------------------------------
// Router: probs[b,t,:] = softmax( tanh(k_flat @ W1 + b1) @ W2 + b2 )
// One 128-thread block per token (B*T = 4096 blocks).
// ---------------------------------------------------------------------------
__global__ __launch_bounds__(128) void router_kernel(
    const float* __restrict__ k, const float* __restrict__ W1,
    const float* __restrict__ b1, const float* __restrict__ W2,
    const float* __restrict__ b2, float* __restrict__ probs) {
  __shared__ float kf[HH * DD];   // flattened key row, 1024 floats
  __shared__ float hid[RH];

  const int token = blockIdx.x;            // b*T + t
  const int b = token >> 10;
  const int t = token & (TT - 1);
  const int j = threadIdx.x;

  // Cooperative load of k_flat[b, t, :]  (k_flat[c] = k[b, c/64, t, c%64])
#pragma unroll
  for (int i = 0; i < (HH * DD) / RH; ++i) {
    const int c = j + RH * i;
    const int h = c >> 6, d = c & 63;
    kf[c] = k[(((size_t)b * HH + h) * TT + t) * DD + d];
  }
  __syncthreads();

  // hidden[j] = tanh(dot(kf, W1[:, j]) + b1[j]); W1 reads are lane-coalesced.
  float acc = b1[j];
  for (int c = 0; c < HH * DD; ++c)
    acc = fmaf(kf[c], W1[(size_t)c * RH + j], acc);
  hid[j] = tanhf(acc);
  __syncthreads();

  // Lanes 0..15 of wave 0: logits + softmax over 16 bins via shfl reductions.
  if (j < NBINS) {
    float lg = b2[j];
#pragma unroll 8
    for (int c = 0; c < RH; ++c)
      lg = fmaf(hid[c], W2[c * NBINS + j], lg);
    float m = lg;
#pragma unroll
    for (int off = 8; off >= 1; off >>= 1)
      m = fmaxf(m, __shfl_xor(m, off, 16));
    const float e = __expf(lg - m);
    float s = e;
#pragma unroll
    for (int off = 8; off >= 1; off >>= 1)
      s += __shfl_xor(s, off, 16);
    probs[(size_t)token * NBINS + j] = e / s;
  }
}

// ---------------------------------------------------------------------------
// Attention: one wave32 per 16-query tile, flash-attention over key tiles.
// All three GEMMs (sim = Pq·Pk^T, S = Q·K^T, O += P·V) use f32 WMMA.
//
// f32 WMMA operand layouts (CDNA5 ISA 7.12.2):
//   A (16x4, MxK):  lane l: M = l%16, float2 = K {4c+2*(l/16), +1}
//   B (4x16, KxN):  lane l: N = l%16, float2 = K {4c+2*(l/16), +1}
//   C/D (16x16):    vgpr r, lane l: M = r + 8*(l/16), N = l%16
// ---------------------------------------------------------------------------
#define VT_STRIDE 18           // padded: conflict-free banks + even fp2 addr
#define PL_STRIDE 17
#define WAVE_LDS (DD * VT_STRIDE + 16 * PL_STRIDE)

__global__ __launch_bounds__(256) void attn_kernel(
    const float* __restrict__ q, const float* __restrict__ k,
    const float* __restrict__ v, const float* __restrict__ probs,
    float* __restrict__ out) {
  __shared__ float lds[8 * WAVE_LDS];

  const int lane = threadIdx.x & 31;
  const int wid = threadIdx.x >> 5;
  const int lh = lane >> 4;            // lane half (0/1)
  const int ln = lane & 15;

  float* Vt = &lds[wid * WAVE_LDS];    // V tile transposed [d][key], stride 18
  float* Pl = Vt + DD * VT_STRIDE;     // P tile [M][kk], stride 17

  const int gw = blockIdx.x * 8 + wid;       // global wave id, 0..4095
  const int qt = gw & 63;                    // query tile
  const int h = (gw >> 6) & (HH - 1);
  const int b = gw >> 10;

  const size_t bh = ((size_t)b * HH + h) * (size_t)TT * DD;

  // Q tile in A-layout registers: 16 chunks of float2 per lane.
  v2f Aq[16];
  {
    const float* qrow = q + bh + (size_t)(qt * 16 + ln) * DD;
#pragma unroll
    for (int c = 0; c < 16; ++c)
      Aq[c] = *(const v2f*)(qrow + 4 * c + 2 * lh);
  }
  // Query bucket-probs in A-layout: 4 chunks.
  v2f Apq[4];
  {
    const float* pr = probs + ((size_t)b * TT + qt * 16 + ln) * NBINS;
#pragma unroll
    for (int c = 0; c < 4; ++c)
      Apq[c] = *(const v2f*)(pr + 4 * c + 2 * lh);
  }

  v8f O[4] = {};                 // 16 x 64 output accumulator (4 N-tiles)
  float mrow[8], lrow[8];
#pragma unroll
  for (int r = 0; r < 8; ++r) { mrow[r] = -__builtin_inff(); lrow[r] = 0.f; }

  const float scale = 0.125f;    // D^-0.5

  for (int kt = 0; kt <= qt; ++kt) {
    // ---- bucket similarity tile: sim = Pq(16x16) @ Pk^T(16x16), 4 WMMAs ----
    v8f sim = {};
    {
      const float* pk = probs + ((size_t)b * TT + kt * 16 + ln) * NBINS;
#pragma unroll
      for (int c = 0; c < 4; ++c) {
        v2f Bpk = *(const v2f*)(pk + 4 * c + 2 * lh);
        sim = __builtin_amdgcn_wmma_f32_16x16x4_f32(
            false, Apq[c], false, Bpk, (short)0, sim, false, false);
      }
    }

    // ---- stage V tile transposed in LDS (coalesced global reads) ----
    {
      const float* vb = v + bh + (size_t)kt * 16 * DD;
#pragma unroll
      for (int i = 0; i < 8; ++i) {
        const int idx = i * 32 + lane;         // = key*64 + d
        const int key = idx >> 6, d = idx & 63;
        Vt[d * VT_STRIDE + key] = vb[idx];
        const int idx2 = idx + 256;
        const int key2 = idx2 >> 6, d2 = idx2 & 63;
        Vt[d2 * VT_STRIDE + key2] = vb[idx2];
        const int idx3 = idx + 512;
        Vt[(idx3 & 63) * VT_STRIDE + (idx3 >> 6)] = vb[idx3];
        const int idx4 = idx + 768;
        Vt[(idx4 & 63) * VT_STRIDE + (idx4 >> 6)] = vb[idx4];
      }
    }

    // ---- score tile: S = Q(16x64) @ K^T(64x16), 16 chained WMMAs ----
    v8f S = {};
    {
      const float* krow = k + bh + (size_t)(kt * 16 + ln) * DD;
#pragma unroll
      for (int c = 0; c < 16; ++c) {
        v2f Bk = *(const v2f*)(krow + 4 * c + 2 * lh);
        S = __builtin_amdgcn_wmma_f32_16x16x4_f32(
            false, Aq[c], false, Bk, (short)0, S, false, false);
      }
    }

    // ---- mask (causal & sim>thresh), scale ----
    const int kcol = kt * 16 + ln;       // this lane's key column (N = ln)
#pragma unroll
    for (int r = 0; r < 8; ++r) {
      const int qi = qt * 16 + r + 8 * lh;
      const bool ok = (kcol <= qi) && (sim[r] > THRESH);
      S[r] = ok ? S[r] * scale : -__builtin_inff();
    }

    // ---- online softmax (row = 16 lanes of one half-wave) ----
    float al[8];
#pragma unroll
    for (int r = 0; r < 8; ++r) {
      float mx = S[r];
#pragma unroll
      for (int off = 8; off >= 1; off >>= 1)
        mx = fmaxf(mx, __shfl_xor(mx, off));
      const float mnew = fmaxf(mrow[r], mx);
      const bool dead = (mnew == -__builtin_inff());
      const float alpha = dead ? 1.0f : __expf(mrow[r] - mnew);
      const float pv = dead ? 0.0f : __expf(S[r] - mnew);
      S[r] = pv;
      float rs = pv;
#pragma unroll
      for (int off = 8; off >= 1; off >>= 1)
        rs += __shfl_xor(rs, off);
      lrow[r] = lrow[r] * alpha + rs;
      mrow[r] = mnew;
      al[r] = alpha;
    }
#pragma unroll
    for (int nt = 0; nt < 4; ++nt)
#pragma unroll
      for (int r = 0; r < 8; ++r)
        O[nt][r] *= al[r];

    // ---- bounce P through LDS: C/D layout -> A layout ----
#pragma unroll
    for (int r = 0; r < 8; ++r)
      Pl[(r + 8 * lh) * PL_STRIDE + ln] = S[r];
    asm volatile("" ::: "memory");   // no compiler reorder; LDS is in-order/wave

    // ---- O += P(16x16) @ V(16x64): 4 K-chunks x 4 N-tiles WMMAs ----
#pragma unroll
    for (int kc = 0; kc < 4; ++kc) {
      v2f Ap;
      const int pa = ln * PL_STRIDE + 4 * kc + 2 * lh;
      Ap[0] = Pl[pa];
      Ap[1] = Pl[pa + 1];
#pragma unroll
      for (int nt = 0; nt < 4; ++nt) {
        v2f Bv;
        const int va = (nt * 16 + ln) * VT_STRIDE + 4 * kc + 2 * lh;
        Bv[0] = Vt[va];
        Bv[1] = Vt[va + 1];
        O[nt] = __builtin_amdgcn_wmma_f32_16x16x4_f32(
            false, Ap, false, Bv, (short)0, O[nt], false, false);
      }
    }
    asm volatile("" ::: "memory");
  }

  // ---- normalize and write out[b,h,qt*16+M, :] ----
  float inv[8];
#pragma unroll
  for (int r = 0; r < 8; ++r)
    inv[r] = (lrow[r] > 0.f) ? 1.0f / lrow[r] : 0.0f;
  float* ob = out + bh;
#pragma unroll
  for (int nt = 0; nt < 4; ++nt)
#pragma unroll
    for (int r = 0; r < 8; ++r) {
      const int M = r + 8 * lh;
      ob[(size_t)(qt * 16 + M) * DD + nt * 16 + ln] = O[nt][r] * inv[r];
    }
}

// ---------------------------------------------------------------------------
extern "C" void kernel_launch(void* const* d_in, const int* in_sizes, int n_in,
                              void* d_out, int out_size, void* d_ws,
                              size_t ws_size, hipStream_t stream) {
  const float* q = (const float*)d_in[0];
  const float* k = (const float*)d_in[1];
  const float* v = (const float*)d_in[2];
  const float* W1 = (const float*)d_in[3];
  const float* b1 = (const float*)d_in[4];
  const float* W2 = (const float*)d_in[5];
  const float* b2 = (const float*)d_in[6];
  float* probs = (float*)d_ws;   // B*T*NBINS floats = 256 KB scratch

  router_kernel<<<BB * TT, RH, 0, stream>>>(k, W1, b1, W2, b2, probs);

  const int nwaves = BB * HH * (TT / 16);  // 4096 waves, 8 per block
  attn_kernel<<<nwaves / 8, 256, 0, stream>>>(q, k, v, probs, (float*)d_out);
}